// GNN_6571299963061
// MI455X (gfx1250) — compile-verified
//
#include <hip/hip_runtime.h>

#define NN 100000
#define NE 625000
#define D  128

typedef float v2f __attribute__((ext_vector_type(2)));
typedef float v8f __attribute__((ext_vector_type(8)));

// ---------------------------------------------------------------------------
// Phase 1: per-node in-degree (float, for mean normalization)
// ---------------------------------------------------------------------------
__global__ __launch_bounds__(256) void sage_deg_kernel(const int* __restrict__ dst,
                                                       float* __restrict__ deg, int E) {
  int e = blockIdx.x * blockDim.x + threadIdx.x;
  if (e < E) atomicAdd(&deg[dst[e]], 1.0f);
}

// ---------------------------------------------------------------------------
// Phase 2: edge-parallel scatter-add: agg[dst] += x[src].
// One wave (32 lanes) per edge, float4 per lane (128 floats / row).
// x rows are 512B-aligned, gathers are L2-resident (x = 51.2MB < 192MB L2).
// ---------------------------------------------------------------------------
__global__ __launch_bounds__(256) void sage_scatter_kernel(const float* __restrict__ x,
                                                           const int* __restrict__ src,
                                                           const int* __restrict__ dst,
                                                           float* __restrict__ agg, int E) {
  long long t = (long long)blockIdx.x * blockDim.x + threadIdx.x;
  int e    = (int)(t >> 5);
  int lane = (int)(t & 31);
  if (e >= E) return;
  int s = src[e];
  int d = dst[e];
  const float4 v = ((const float4*)(x + (long long)s * D))[lane];
  float* a = agg + (long long)d * D + lane * 4;
  atomicAdd(a + 0, v.x);
  atomicAdd(a + 1, v.y);
  atomicAdd(a + 2, v.z);
  atomicAdd(a + 3, v.w);
}

// ---------------------------------------------------------------------------
// Phase 3: fused SAGE layer via V_WMMA_F32_16X16X4_F32 (exact fp32):
//   out[16 rows] = (agg * 1/max(deg,1)) @ Wl^T + b + xin @ Wr^T  [+ ReLU]
// One wave per 16-row tile; 8 col-tiles (128 cols) of v8f accumulators.
// Two K=128 passes (agg/Wl then xin/Wr) accumulate into the same C.
//
// Fragment layouts (ISA 7.12.2, wave32):
//   A 16x4 f32 : lane m(0-15): v0=A[m][k0],   v1=A[m][k0+1]
//                lane m+16   : v0=A[m][k0+2], v1=A[m][k0+3]
//   B 4x16 f32 : lane c(0-15): v0=B[k0][c],   v1=B[k0+1][c]
//                lane c+16   : v0=B[k0+2][c], v1=B[k0+3][c]   (B[k][c]=W[c][k])
//   C/D 16x16  : lane c(0-15) vgpr r = D[r][c]; lane c+16 vgpr r = D[r+8][c]
// ---------------------------------------------------------------------------
__global__ __launch_bounds__(256) void sage_wmma_kernel(const float* __restrict__ agg,
                                                        const float* __restrict__ deg,
                                                        const float* __restrict__ xin,
                                                        const float* __restrict__ Wl,
                                                        const float* __restrict__ Wr,
                                                        const float* __restrict__ bias,
                                                        float* __restrict__ out,
                                                        int n, int applyRelu) {
  const int lane = threadIdx.x & 31;
  const int wave = threadIdx.x >> 5;
  const int tile = blockIdx.x * 8 + wave;   // 16-row output tile
  const int nTiles = n >> 4;                // n % 16 == 0
  if (tile >= nTiles) return;               // wave-uniform: EXEC stays all-1s
  const int row0 = tile * 16;

  const int m  = lane & 15;                 // A-row / B-col / D-col within tile
  const int kh = (lane >> 4) * 2;           // k sub-offset for hi half-wave

  // mean normalization factor for this A row (isolated nodes -> divide by 1)
  const float rdeg = 1.0f / fmaxf(deg[row0 + m], 1.0f);

  v8f zero = {};
  v8f acc[8];
#pragma unroll
  for (int i = 0; i < 8; ++i) acc[i] = zero;

  for (int pass = 0; pass < 2; ++pass) {
    const float* A     = pass ? xin : agg;
    const float* W     = pass ? Wr : Wl;
    const float  scale = pass ? 1.0f : rdeg;
    const float* arow  = A + (long long)(row0 + m) * D;
#pragma unroll 2
    for (int k0 = 0; k0 < D; k0 += 4) {
      v2f a;
      a.x = arow[k0 + kh + 0] * scale;
      a.y = arow[k0 + kh + 1] * scale;
#pragma unroll
      for (int ct = 0; ct < 8; ++ct) {
        const float* wrow = W + (long long)(ct * 16 + m) * D;  // W[c][.]
        v2f b;
        b.x = wrow[k0 + kh + 0];
        b.y = wrow[k0 + kh + 1];
        acc[ct] = __builtin_amdgcn_wmma_f32_16x16x4_f32(
            /*neg_a=*/false, a, /*neg_b=*/false, b,
            /*c_mod=*/(short)0, acc[ct], /*reuse_a=*/false, /*reuse_b=*/false);
      }
    }
  }

  // Epilogue: bias + optional ReLU + store (contiguous 64B per half-wave row)
  const int rhalf = (lane >> 4) * 8;
#pragma unroll
  for (int ct = 0; ct < 8; ++ct) {
    const float bc = bias[ct * 16 + m];
#pragma unroll
    for (int r = 0; r < 8; ++r) {
      float v = acc[ct][r] + bc;
      if (applyRelu) v = fmaxf(v, 0.0f);
      out[(long long)(row0 + r + rhalf) * D + ct * 16 + m] = v;
    }
  }
}

// ---------------------------------------------------------------------------
// Host-side launch (graph-capture safe: only kernels + hipMemsetAsync)
// ---------------------------------------------------------------------------
extern "C" void kernel_launch(void* const* d_in, const int* in_sizes, int n_in,
                              void* d_out, int out_size, void* d_ws, size_t ws_size,
                              hipStream_t stream) {
  (void)in_sizes; (void)n_in; (void)out_size; (void)ws_size;

  const float* x   = (const float*)d_in[0];
  const int*   ei  = (const int*)d_in[1];   // [2, E] int32: row0=src, row1=dst
  const float* W1l = (const float*)d_in[2];
  const float* b1  = (const float*)d_in[3];
  const float* W1r = (const float*)d_in[4];
  const float* W2l = (const float*)d_in[5];
  const float* b2  = (const float*)d_in[6];
  const float* W2r = (const float*)d_in[7];
  float*       out = (float*)d_out;

  const int* src = ei;
  const int* dst = ei + NE;

  // workspace layout
  char*  ws  = (char*)d_ws;
  float* deg = (float*)ws;                                   // N floats   (0.4 MB)
  float* agg = (float*)(ws + (1ull << 20));                  // N*D floats (51.2 MB)
  float* h   = (float*)(ws + (1ull << 20) + (size_t)NN * D * sizeof(float));

  const int degBlocks     = (NE + 255) / 256;
  const int scatterBlocks = (int)(((long long)NE * 32 + 255) / 256);
  const int gemmBlocks    = ((NN / 16) + 7) / 8;             // 8 waves/block

  // ---- layer 1 ----
  hipMemsetAsync(deg, 0, (size_t)NN * sizeof(float), stream);
  hipMemsetAsync(agg, 0, (size_t)NN * D * sizeof(float), stream);
  sage_deg_kernel<<<degBlocks, 256, 0, stream>>>(dst, deg, NE);
  sage_scatter_kernel<<<scatterBlocks, 256, 0, stream>>>(x, src, dst, agg, NE);
  sage_wmma_kernel<<<gemmBlocks, 256, 0, stream>>>(agg, deg, x, W1l, W1r, b1, h,
                                                   NN, /*relu=*/1);

  // ---- layer 2 (same graph: deg reused) ----
  hipMemsetAsync(agg, 0, (size_t)NN * D * sizeof(float), stream);
  sage_scatter_kernel<<<scatterBlocks, 256, 0, stream>>>(h, src, dst, agg, NE);
  sage_wmma_kernel<<<gemmBlocks, 256, 0, stream>>>(agg, deg, h, W2l, W2r, b2, out,
                                                   NN, /*relu=*/0);
}